// QSAL_pennylane_32598801776753
// MI455X (gfx1250) — compile-verified
//
#include <hip/hip_runtime.h>
#include <hip/hip_bf16.h>

// ---------------------------------------------------------------------------
// QSAL quantum-self-attention on MI455X (gfx1250).
// Per-token 8-qubit statevector sim expressed as chained 16x16 fp32 WMMA
// matmuls (state held as 16x16 complex matrix in per-wave LDS). CNOT rings
// are folded into a prefix-XOR index permutation (pure LDS shuffle).
// ---------------------------------------------------------------------------

#define N_Q     8
#define SEQ     64
#define BATCH   256
#define D_FEAT  40
#define NSTATE  256                      // 2^8 amplitudes
#define WAVES_PER_BLOCK 4
#define UNITS   (3 * BATCH * SEQ)        // (Q,K,V) x tokens

typedef __attribute__((ext_vector_type(2))) float v2f;
typedef __attribute__((ext_vector_type(8))) float v8f;

struct C2x2 { float r00, r01, r10, r11, i00, i01, i10, i11; };
struct R2x2 { float r00, r01, r10, r11; };

// D(16x16) += (NEGA? -A : A)(16x16) * B(16x16), B optionally accessed
// transposed. K=16 done as 4 chained V_WMMA_F32_16X16X4_F32.
// A-frag (ISA 16x4 layout): row = lane&15, VGPR0/1 hold K = 2*half + {0,1}.
// B-frag: col = lane&15, rows K = 2*half + {0,1} per k-chunk.
template <bool NEGA, bool BTRANS>
static __device__ __forceinline__ v8f mm16(const float* A, const float* B,
                                           v8f acc, int lane) {
  const int half = (lane >> 4) & 1;
  const int mn   = lane & 15;
#pragma unroll
  for (int kc = 0; kc < 4; ++kc) {
    const int k0 = kc * 4 + half * 2;
    v2f a;
    a.x = A[mn * 16 + k0];
    a.y = A[mn * 16 + k0 + 1];
    if (NEGA) { a.x = -a.x; a.y = -a.y; }
    v2f b;
    if (BTRANS) {
      b.x = B[mn * 16 + k0];          // B^T: element [k][n] = B[n][k]
      b.y = B[mn * 16 + k0 + 1];
    } else {
      b.x = B[k0 * 16 + mn];
      b.y = B[(k0 + 1) * 16 + mn];
    }
    acc = __builtin_amdgcn_wmma_f32_16x16x4_f32(false, a, false, b, (short)0,
                                                acc, false, false);
  }
  return acc;
}

// Scatter C/D fragment (rows 8*half + r, col lane&15) back to row-major LDS.
static __device__ __forceinline__ void storeD(float* D, v8f d, int lane) {
  const int half = (lane >> 4) & 1;
  const int n    = lane & 15;
#pragma unroll
  for (int r = 0; r < 8; ++r) D[(half * 8 + r) * 16 + n] = d[r];
}

static __device__ __forceinline__ void sel(const C2x2& g, int ri, int ci,
                                           float& ar, float& ai) {
  float r0 = ci ? g.r01 : g.r00, r1 = ci ? g.r11 : g.r10;
  float i0 = ci ? g.i01 : g.i00, i1 = ci ? g.i11 : g.i10;
  ar = ri ? r1 : r0;
  ai = ri ? i1 : i0;
}

// G = g3 (x) g2 (x) g1 (x) g0  (complex 16x16), element [r][c] = prod over
// qubit j of g[j][r_j][c_j]. 8 elements per lane.
static __device__ void buildG4C(float* Gre, float* Gim, const C2x2 g[4],
                                int lane) {
#pragma unroll
  for (int t = 0; t < 8; ++t) {
    int e = lane * 8 + t;
    int r = e >> 4, c = e & 15;
    float pr = 1.f, pi = 0.f;
#pragma unroll
    for (int j = 0; j < 4; ++j) {
      float ar, ai;
      sel(g[j], (r >> j) & 1, (c >> j) & 1, ar, ai);
      float nr = pr * ar - pi * ai;
      float ni = pr * ai + pi * ar;
      pr = nr; pi = ni;
    }
    Gre[e] = pr;
    Gim[e] = pi;
  }
}

static __device__ void buildG4R(float* Gre, const R2x2 g[4], int lane) {
#pragma unroll
  for (int t = 0; t < 8; ++t) {
    int e = lane * 8 + t;
    int r = e >> 4, c = e & 15;
    float pr = 1.f;
#pragma unroll
    for (int j = 0; j < 4; ++j) {
      int ri = (r >> j) & 1, ci = (c >> j) & 1;
      const R2x2& q = g[j];
      float v0 = ci ? q.r01 : q.r00, v1 = ci ? q.r11 : q.r10;
      pr *= ri ? v1 : v0;
    }
    Gre[e] = pr;
  }
}

// D = G * M  (complex): transforms high-nibble qubits 4..7.
static __device__ void applyLeftC(const float* Gre, const float* Gim,
                                  const float* Mre, const float* Mim,
                                  float* Dre, float* Dim, int lane) {
  v8f ar = {};
  ar = mm16<false, false>(Gre, Mre, ar, lane);
  ar = mm16<true,  false>(Gim, Mim, ar, lane);
  v8f ai = {};
  ai = mm16<false, false>(Gre, Mim, ai, lane);
  ai = mm16<false, false>(Gim, Mre, ai, lane);
  storeD(Dre, ar, lane);
  storeD(Dim, ai, lane);
}

// D = M * G^T (complex): transforms low-nibble qubits 0..3.
static __device__ void applyRightC(const float* Mre, const float* Mim,
                                   const float* Gre, const float* Gim,
                                   float* Dre, float* Dim, int lane) {
  v8f ar = {};
  ar = mm16<false, true>(Mre, Gre, ar, lane);
  ar = mm16<true,  true>(Mim, Gim, ar, lane);
  v8f ai = {};
  ai = mm16<false, true>(Mre, Gim, ai, lane);
  ai = mm16<false, true>(Mim, Gre, ai, lane);
  storeD(Dre, ar, lane);
  storeD(Dim, ai, lane);
}

static __device__ void applyLeftR(const float* G, const float* Mre,
                                  const float* Mim, float* Dre, float* Dim,
                                  int lane) {
  v8f ar = {};
  ar = mm16<false, false>(G, Mre, ar, lane);
  v8f ai = {};
  ai = mm16<false, false>(G, Mim, ai, lane);
  storeD(Dre, ar, lane);
  storeD(Dim, ai, lane);
}

static __device__ void applyRightR(const float* Mre, const float* Mim,
                                   const float* G, float* Dre, float* Dim,
                                   int lane) {
  v8f ar = {};
  ar = mm16<false, true>(Mre, G, ar, lane);
  v8f ai = {};
  ai = mm16<false, true>(Mim, G, ai, lane);
  storeD(Dre, ar, lane);
  storeD(Dim, ai, lane);
}

// CNOT ring (0->1,1->2,...,7->0) as an index permutation:
// new bit k (k>=1) = prefix-xor(b0..bk); new bit 0 = b0 ^ parity(all 8 bits).
static __device__ void permuteRing(const float* Sre, const float* Sim,
                                   float* Dre, float* Dim, int lane) {
#pragma unroll
  for (int t = 0; t < 8; ++t) {
    int f = lane * 8 + t;
    int x = f;
    x ^= x << 1;
    x ^= x << 2;
    x ^= x << 4;
    x &= 255;                              // bit k of x = prefix-xor p_k
    int nf = (x & 0xFE) | ((f ^ (x >> 7)) & 1);
    Dre[nf] = Sre[f];
    Dim[nf] = Sim[f];
  }
}

// U = RY(ty) * RX(tx) fused single-qubit gate.
static __device__ C2x2 fuseRxRy(float tx, float ty) {
  float c0 = __cosf(0.5f * tx), s0 = __sinf(0.5f * tx);
  float c1 = __cosf(0.5f * ty), s1 = __sinf(0.5f * ty);
  C2x2 g;
  g.r00 = c1 * c0;  g.r01 = -s1 * c0;  g.r10 = s1 * c0;   g.r11 = c1 * c0;
  g.i00 = s1 * s0;  g.i01 = -c1 * s0;  g.i10 = -c1 * s0;  g.i11 = -s1 * s0;
  return g;
}

static __device__ R2x2 ryGate(float t) {
  float c = __cosf(0.5f * t), s = __sinf(0.5f * t);
  R2x2 g;
  g.r00 = c; g.r01 = -s; g.r10 = s; g.r11 = c;
  return g;
}

#define SWAP_ST()                                        \
  { float* tp = Sre; Sre = Tre; Tre = tp;                \
    tp = Sim; Sim = Tim; Tim = tp; }

__global__ __launch_bounds__(WAVES_PER_BLOCK * 32)
void qsal_circuit_kernel(const float* __restrict__ x,
                         const float* __restrict__ pQ,
                         const float* __restrict__ pK,
                         const float* __restrict__ pV,
                         float* __restrict__ qkv) {
  __shared__ float lds[WAVES_PER_BLOCK * 6 * NSTATE];
  const int lane = threadIdx.x & 31;
  const int wv   = threadIdx.x >> 5;
  float* base = lds + wv * 6 * NSTATE;
  float* Sre = base;
  float* Sim = base + NSTATE;
  float* Tre = base + 2 * NSTATE;
  float* Tim = base + 3 * NSTATE;
  float* Gre = base + 4 * NSTATE;
  float* Gim = base + 5 * NSTATE;

  const int u = blockIdx.x * WAVES_PER_BLOCK + wv;
  if (u >= UNITS) return;
  const int p  = u / (BATCH * SEQ);          // 0:Q 1:K 2:V
  const int tk = u % (BATCH * SEQ);
  const int b  = tk / SEQ;
  const int s  = tk % SEQ;
  const float* xa = x + (size_t)(b * SEQ + s) * D_FEAT;
  const float* wa = (p == 0 ? pQ : (p == 1 ? pK : pV)) + (size_t)s * D_FEAT;

  // |0...0>
#pragma unroll
  for (int t = 0; t < 8; ++t) {
    int f = lane * 8 + t;
    Sre[f] = 0.f;
    Sim[f] = 0.f;
  }
  if (lane == 0) Sre[0] = 1.f;

  for (int phase = 0; phase < 2; ++phase) {
    const float* a = phase ? wa : xa;
    // Per-qubit RX,RY on qubits 0..3 fused into complex 16x16 (low nibble).
    {
      C2x2 g[4];
#pragma unroll
      for (int j = 0; j < 4; ++j) g[j] = fuseRxRy(a[2 * j], a[2 * j + 1]);
      buildG4C(Gre, Gim, g, lane);
      applyRightC(Sre, Sim, Gre, Gim, Tre, Tim, lane);
      SWAP_ST();
    }
    // Qubits 4..7 (high nibble).
    {
      C2x2 g[4];
#pragma unroll
      for (int j = 0; j < 4; ++j) g[j] = fuseRxRy(a[8 + 2 * j], a[9 + 2 * j]);
      buildG4C(Gre, Gim, g, lane);
      applyLeftC(Gre, Gim, Sre, Sim, Tre, Tim, lane);
      SWAP_ST();
    }
    // 3 entangling layers: CNOT ring, then RY on every qubit.
    for (int L = 0; L < 3; ++L) {
      permuteRing(Sre, Sim, Tre, Tim, lane);
      SWAP_ST();
      {
        R2x2 h[4];
#pragma unroll
        for (int j = 0; j < 4; ++j) h[j] = ryGate(a[16 + 8 * L + j]);
        buildG4R(Gre, h, lane);
        applyRightR(Sre, Sim, Gre, Tre, Tim, lane);
        SWAP_ST();
      }
      {
        R2x2 h[4];
#pragma unroll
        for (int j = 0; j < 4; ++j) h[j] = ryGate(a[16 + 8 * L + 4 + j]);
        buildG4R(Gre, h, lane);
        applyLeftR(Gre, Sre, Sim, Tre, Tim, lane);
        SWAP_ST();
      }
    }
  }

  // Expectation: Z0 (sign by bit 0) for Q/K, Z^(x)8 (sign by parity) for V.
  float acc = 0.f;
#pragma unroll
  for (int t = 0; t < 8; ++t) {
    int f = lane * 8 + t;
    float pr = Sre[f] * Sre[f] + Sim[f] * Sim[f];
    int sgn = (p == 2) ? (__popc(f) & 1) : (f & 1);
    acc += sgn ? -pr : pr;
  }
#pragma unroll
  for (int off = 16; off; off >>= 1) acc += __shfl_xor(acc, off, 32);
  if (lane == 0) qkv[u] = acc;
}

// Gaussian-kernel attention + residual broadcast.
__global__ void qsal_attn_kernel(const float* __restrict__ x,
                                 const float* __restrict__ qkv,
                                 float* __restrict__ out) {
  int tid = blockIdx.x * blockDim.x + threadIdx.x;  // over BATCH*SEQ
  if (tid >= BATCH * SEQ) return;
  int b = tid / SEQ;
  const float* Q = qkv;
  const float* K = qkv + BATCH * SEQ;
  const float* V = qkv + 2 * BATCH * SEQ;
  float q = Q[tid];
  float num = 0.f, den = 0.f;
#pragma unroll 4
  for (int m = 0; m < SEQ; ++m) {
    float d = q - K[b * SEQ + m];
    float al = __expf(-d * d);
    num += al * V[b * SEQ + m];
    den += al;
  }
  float att = num / den;
  const float* xp = x + (size_t)tid * D_FEAT;
  float* op = out + (size_t)tid * D_FEAT;
#pragma unroll
  for (int d = 0; d < D_FEAT; ++d) op[d] = xp[d] + att;
}

extern "C" void kernel_launch(void* const* d_in, const int* in_sizes, int n_in,
                              void* d_out, int out_size, void* d_ws,
                              size_t ws_size, hipStream_t stream) {
  (void)in_sizes; (void)n_in; (void)out_size; (void)ws_size;
  const float* x  = (const float*)d_in[0];
  const float* pQ = (const float*)d_in[1];
  const float* pK = (const float*)d_in[2];
  const float* pV = (const float*)d_in[3];
  float* out = (float*)d_out;
  float* qkv = (float*)d_ws;  // 3 * BATCH * SEQ floats (Q | K | V)

  dim3 blk1(WAVES_PER_BLOCK * 32);
  dim3 grd1(UNITS / WAVES_PER_BLOCK);
  qsal_circuit_kernel<<<grd1, blk1, 0, stream>>>(x, pQ, pK, pV, qkv);

  dim3 blk2(256);
  dim3 grd2((BATCH * SEQ + 255) / 256);
  qsal_attn_kernel<<<grd2, blk2, 0, stream>>>(x, qkv, out);
}